// IAF_30743375904880
// MI455X (gfx1250) — compile-verified
//
#include <hip/hip_runtime.h>
#include <cstdint>

// Problem geometry (from reference): data (B*T, F) = (32*1024, 1024) f32.
#define BB 32
#define TT 1024
#define FF 1024

// Pipeline geometry.
#define TS        8                    // timesteps per tile
#define NBUF      8                    // ring depth (tiles in flight per wave)
#define NT        (TT / TS)            // 128 tiles
#define WAVE_COLS 32                   // columns owned by one wave (wave32)
#define THREADS   64                   // 2 waves / block
#define BLK_COLS  (THREADS)            // 64 columns per block
#define NBLOCKS   (BB * (FF / BLK_COLS))  // 32 * 16 = 512

// One async b128 DMA: global -> LDS. VDST VGPR carries the LDS byte address,
// VADDR pair carries the 64-bit global address (GV mode, SADDR=off).
__device__ __forceinline__ void async_ld_b128(const float* g, unsigned lds_byte) {
  asm volatile("global_load_async_to_lds_b128 %0, %1, off"
               :: "v"(lds_byte), "v"(g)
               : "memory");
}

__device__ __forceinline__ void wait_async_le_14() {
#if __has_builtin(__builtin_amdgcn_s_wait_asynccnt)
  __builtin_amdgcn_s_wait_asynccnt(14);   // (NBUF-1)*2 -> oldest tile retired
#else
  asm volatile("s_wait_asynccnt 14" ::: "memory");
#endif
}

__device__ __forceinline__ void wait_ds_zero() {
#if __has_builtin(__builtin_amdgcn_s_wait_dscnt)
  __builtin_amdgcn_s_wait_dscnt(0);
#else
  asm volatile("s_wait_dscnt 0" ::: "memory");
#endif
}

__global__ __launch_bounds__(THREADS)
void iaf_scan_kernel(const float* __restrict__ x, float* __restrict__ out) {
  // Per-wave-private ring: [wave][buf][row(TS)][col(32)] floats.
  __shared__ float lds[2 * NBUF * TS * WAVE_COLS];   // 4096 f32 = 16 KB

  const int tid = threadIdx.x;
  const int w   = tid >> 5;          // wave id in block (0..1)
  const int l   = tid & 31;          // lane id

  const int b   = blockIdx.x >> 4;                 // FF/BLK_COLS = 16 blocks per batch
  const int f0  = (blockIdx.x & 15) * BLK_COLS;    // block's first column
  const int cw  = f0 + w * WAVE_COLS;              // wave's first column

  // Global bases (float indices): element (t, c) of this wave = gwave + t*FF + c
  const float* gwave = x + (size_t)b * TT * FF + cw;
  float*       ocol  = out + (size_t)b * TT * FF + f0 + tid;  // this thread's column

  // LDS byte base of this wave's ring (low 32 bits of a flat LDS pointer = LDS offset)
  const unsigned lds_wave =
      (unsigned)(uintptr_t)(&lds[w * NBUF * TS * WAVE_COLS]);

  // Lane -> two b128 chunks of each tile:
  //   chunk c = l     -> row r0 = l>>3,   cols [(l&7)*4 .. +3]
  //   chunk c = l+32  -> row r0+4,        same cols
  const int r0 = l >> 3;
  const int c0 = (l & 7) * 4;        // float offset within a 32-col row

  auto issue_tile = [&](int t0, int buf) {
    const unsigned base = lds_wave + (unsigned)(buf * TS * WAVE_COLS * 4);
    const float* g0 = gwave + (size_t)(t0 + r0) * FF + c0;
    const float* g1 = gwave + (size_t)(t0 + r0 + 4) * FF + c0;
    async_ld_b128(g0, base + (unsigned)(((r0    ) * WAVE_COLS + c0) * 4));
    async_ld_b128(g1, base + (unsigned)(((r0 + 4) * WAVE_COLS + c0) * 4));
  };

  // Prologue: fill the ring (16 DMAs in flight per wave).
#pragma unroll
  for (int k = 0; k < NBUF; ++k) issue_tile(k * TS, k);

  float state = 0.0f, act = 0.0f;

  for (int k = 0; k < NT; ++k) {
    wait_async_le_14();              // oldest tile (k) fully landed in LDS
    const int buf = k & (NBUF - 1);
    const float* tile =
        &lds[w * NBUF * TS * WAVE_COLS + buf * TS * WAVE_COLS + l];

    float xs[TS];
#pragma unroll
    for (int tl = 0; tl < TS; ++tl) xs[tl] = tile[tl * WAVE_COLS];  // ds_load_b32

#pragma unroll
    for (int tl = 0; tl < TS; ++tl) {
      state = state + xs[tl] - act;              // membrane update (THRESHOLD=1)
      state = fmaxf(state, -1.0f);               // lower-bound clamp
      act   = fmaxf(floorf(state), 0.0f);        // (v>0)*floor(v)
      __builtin_nontemporal_store(act, ocol + (size_t)(k * TS + tl) * FF);
    }

    // Refill this ring slot. Past the end, re-load the last tile into the
    // (never-again-consumed) slot so ASYNCcnt stays constant and the fixed
    // s_wait_asynccnt immediate keeps retiring exactly the oldest tile.
    const int t0n = (k + NBUF < NT) ? (k + NBUF) * TS : (NT - 1) * TS;
    wait_ds_zero();                  // ds_loads retired before DMA overwrite
    issue_tile(t0n, buf);
  }
}

extern "C" void kernel_launch(void* const* d_in, const int* in_sizes, int n_in,
                              void* d_out, int out_size, void* d_ws, size_t ws_size,
                              hipStream_t stream) {
  const float* x = (const float*)d_in[0];
  float* out = (float*)d_out;
  (void)in_sizes; (void)n_in; (void)out_size; (void)d_ws; (void)ws_size;
  hipLaunchKernelGGL(iaf_scan_kernel, dim3(NBLOCKS), dim3(THREADS), 0, stream,
                     x, out);
}